// SelfTrainingLoss_19292993093663
// MI455X (gfx1250) — compile-verified
//
#include <hip/hip_runtime.h>
#include <stdint.h>

// ---------------------------------------------------------------------------
// SelfTrainingLoss for pred [4,19,512,1024] f32 on MI455X (gfx1250).
//
// Memory-bound: 159 MB pred stream -> ~7 us floor at 23.3 TB/s. No GEMM, so
// no WMMA; the CDNA5 path exercised here is the async global->LDS data mover
// (GLOBAL_LOAD_ASYNC_TO_LDS_B128, ASYNCcnt) staging 19 channel rows per block
// into LDS for the two-pass stable softmax, plus wave32-sized blocks.
// Per-(image,class) top-k is done with an exact 2-level radix select on the
// float bit pattern of max_prob (monotone for positive floats).
// ---------------------------------------------------------------------------

constexpr int   B_ = 4, C_ = 19, H_ = 512, W_ = 1024;
constexpr int   N_ = H_ * W_;            // 524288 = 2^19 pixels per image
constexpr int   TOTAL_ = B_ * N_;        // 2097152 pixels
constexpr int   BC_ = B_ * C_;           // 76 (image,class) pairs
constexpr int   NBINS1 = 10241;          // coarse bins: (bits-0x3D000000)>>12, p in [1/19,1]
constexpr int   NBINS2 = 4096;           // fine bins: low 12 bits
constexpr unsigned PBASE = 0x3D000000u;  // bits of 2^-5 (< 1/19, so no underflow)
constexpr double FIXSCALE = 68719476736.0; // 2^36 fixed point for deterministic atomics

constexpr int TPB   = 256;               // 8 wave32 waves
constexpr int PIXPT = 4;                 // pixels per thread (float4 path)
constexpr int PIXPB = TPB * PIXPT;       // 1024 pixels per block
constexpr int NBLK_PIX = TOTAL_ / PIXPB; // 2048 blocks

// workspace layout (~23 MB)
constexpr size_t OFF_ACC   = 0;                                         // u64 accumulator
constexpr size_t OFF_HIST1 = 16;
constexpr size_t OFF_HIST2 = OFF_HIST1 + (size_t)BC_ * NBINS1 * 4;
constexpr size_t ZERO_BYTES= OFF_HIST2 + (size_t)BC_ * NBINS2 * 4;      // memset range
constexpr size_t OFF_T1    = ZERO_BYTES;
constexpr size_t OFF_KIN   = OFF_T1  + (size_t)BC_ * 4;
constexpr size_t OFF_THR   = OFF_KIN + (size_t)BC_ * 4;
constexpr size_t OFF_MAXP  = (OFF_THR + (size_t)BC_ * 4 + 15) & ~(size_t)15;
constexpr size_t OFF_NLL   = OFF_MAXP + (size_t)TOTAL_ * 4;
constexpr size_t OFF_PS    = OFF_NLL  + (size_t)TOTAL_ * 4;

// --- CDNA5 async global->LDS copy (ASYNCcnt tracked) -----------------------
__device__ __forceinline__ void async_copy_b128(unsigned lds_off, const void* gptr) {
  unsigned long long ga = (unsigned long long)gptr;
  asm volatile("global_load_async_to_lds_b128 %0, %1, off"
               :: "v"(lds_off), "v"(ga) : "memory");
}
__device__ __forceinline__ void wait_async0() {
  asm volatile("s_wait_asynccnt 0x0" ::: "memory");
}

// ---------------------------------------------------------------------------
// Kernel A: per-pixel softmax stats + coarse histogram.
//   max_prob = 1/s, nll = log(s), pseudo = argmax (first-wins ties like jnp).
// 19 channel rows (1024 floats each) staged into LDS via async copies so the
// two passes (max, then sum-exp) read LDS, not HBM/VGPR-hoard.
// ---------------------------------------------------------------------------
__global__ __launch_bounds__(TPB) void stats_kernel(const float* __restrict__ pred,
                                                    float* __restrict__ maxp,
                                                    float* __restrict__ nllp,
                                                    unsigned* __restrict__ pseudo4,
                                                    unsigned* __restrict__ hist1) {
  __shared__ __align__(16) float smem[C_ * PIXPB];   // 19*1024*4 = 76 KB (<320 KB/WGP)
  const int t     = threadIdx.x;
  const int gpix0 = blockIdx.x * PIXPB;              // all 1024 px share one image
  const int b     = gpix0 >> 19;
  const int pix0  = gpix0 & (N_ - 1);
  const float* base = pred + (size_t)b * C_ * N_ + pix0;

#pragma unroll
  for (int c = 0; c < C_; ++c) {
    unsigned loff = (unsigned)(uintptr_t)(&smem[c * PIXPB + t * 4]);
    async_copy_b128(loff, base + (size_t)c * N_ + t * 4);   // 16B/lane, coalesced
  }
  wait_async0();          // this wave's async writes to LDS done
  __syncthreads();        // make all waves' rows visible

  // pass 1: max + argmax
  float4 x0 = *(const float4*)&smem[t * 4];
  float m0 = x0.x, m1 = x0.y, m2 = x0.z, m3 = x0.w;
  int   a0 = 0, a1 = 0, a2 = 0, a3 = 0;
#pragma unroll
  for (int c = 1; c < C_; ++c) {
    float4 v = *(const float4*)&smem[c * PIXPB + t * 4];
    if (v.x > m0) { m0 = v.x; a0 = c; }
    if (v.y > m1) { m1 = v.y; a1 = c; }
    if (v.z > m2) { m2 = v.z; a2 = c; }
    if (v.w > m3) { m3 = v.w; a3 = c; }
  }
  // pass 2: s = sum exp(x - m)  (s >= 1 since the max term is exp(0)=1)
  float s0 = 0.f, s1 = 0.f, s2 = 0.f, s3 = 0.f;
#pragma unroll
  for (int c = 0; c < C_; ++c) {
    float4 v = *(const float4*)&smem[c * PIXPB + t * 4];
    s0 += __expf(v.x - m0); s1 += __expf(v.y - m1);
    s2 += __expf(v.z - m2); s3 += __expf(v.w - m3);
  }
  float p0 = 1.0f / s0, p1 = 1.0f / s1, p2 = 1.0f / s2, p3 = 1.0f / s3;
  float l0 = __logf(s0), l1 = __logf(s1), l2 = __logf(s2), l3 = __logf(s3);

  const int gp = gpix0 + t * 4;
  *(float4*)&maxp[gp] = make_float4(p0, p1, p2, p3);
  *(float4*)&nllp[gp] = make_float4(l0, l1, l2, l3);
  pseudo4[gp >> 2] = (unsigned)a0 | ((unsigned)a1 << 8) |
                     ((unsigned)a2 << 16) | ((unsigned)a3 << 24);

  // coarse radix histogram: float-bit key is monotone in p (p in [1/19, 1])
  unsigned u;
  u = __float_as_uint(p0); atomicAdd(&hist1[(size_t)(b*C_+a0)*NBINS1 + ((u-PBASE)>>12)], 1u);
  u = __float_as_uint(p1); atomicAdd(&hist1[(size_t)(b*C_+a1)*NBINS1 + ((u-PBASE)>>12)], 1u);
  u = __float_as_uint(p2); atomicAdd(&hist1[(size_t)(b*C_+a2)*NBINS1 + ((u-PBASE)>>12)], 1u);
  u = __float_as_uint(p3); atomicAdd(&hist1[(size_t)(b*C_+a3)*NBINS1 + ((u-PBASE)>>12)], 1u);
}

// ---------------------------------------------------------------------------
// Kernel B: per (b,c) find coarse bin t1 containing the k-th largest value.
// k = floor(count * 0.66f) exactly as jnp (f32 multiply + floor).
// ---------------------------------------------------------------------------
__global__ __launch_bounds__(TPB) void coarse_select(const unsigned* __restrict__ hist1,
                                                     int* __restrict__ t1,
                                                     unsigned* __restrict__ kin) {
  const int bc = blockIdx.x;
  const unsigned* h = hist1 + (size_t)bc * NBINS1;
  __shared__ unsigned csum[TPB];
  constexpr int CH = (NBINS1 + TPB - 1) / TPB;   // 41
  const int lo = threadIdx.x * CH;
  const int hi = min(lo + CH, NBINS1);
  unsigned s = 0;
  for (int i = lo; i < hi; ++i) s += h[i];
  csum[threadIdx.x] = s;
  __syncthreads();
  if (threadIdx.x == 0) {
    unsigned total = 0;
    for (int i = 0; i < TPB; ++i) total += csum[i];
    unsigned k = (unsigned)floorf((float)total * 0.66f);
    if (k == 0) { t1[bc] = -1; kin[bc] = 0; return; }
    unsigned cum = 0; int ch = TPB - 1;
    while (cum + csum[ch] < k) { cum += csum[ch]; --ch; }
    const int lo2 = ch * CH;
    int bin = min(lo2 + CH, NBINS1) - 1;
    while (cum + h[bin] < k) { cum += h[bin]; --bin; }
    t1[bc]  = bin;           // coarse bin of the k-th value
    kin[bc] = k - cum;       // how many still to take inside that bin (>=1)
  }
}

// ---------------------------------------------------------------------------
// Kernel C: fine histogram of the low 12 bits, only for pixels that landed in
// their class's coarse threshold bin (few pixels -> few atomics).
// ---------------------------------------------------------------------------
__global__ __launch_bounds__(TPB) void fine_hist(const float* __restrict__ maxp,
                                                 const unsigned* __restrict__ pseudo4,
                                                 const int* __restrict__ t1,
                                                 unsigned* __restrict__ hist2) {
  const int gp = (blockIdx.x * TPB + threadIdx.x) * 4;
  const int b  = gp >> 19;
  float4 p = *(const float4*)&maxp[gp];
  const unsigned ps = pseudo4[gp >> 2];
  float pv[4] = {p.x, p.y, p.z, p.w};
#pragma unroll
  for (int j = 0; j < 4; ++j) {
    unsigned u = __float_as_uint(pv[j]);
    int bc = b * C_ + ((ps >> (8 * j)) & 0xFF);
    if ((int)((u - PBASE) >> 12) == t1[bc])
      atomicAdd(&hist2[(size_t)bc * NBINS2 + (u & 0xFFFu)], 1u);
  }
}

// ---------------------------------------------------------------------------
// Kernel D: exact threshold bits T per (b,c) + tie contribution.
// Ties share an identical max_prob hence identical nll = -log(p_T), so the
// reference's stable-sort tie choice is irrelevant to the scalar loss:
// contribution = rEq * (-log p_T), added only if p_T <= 0.9 (else those
// pixels are already counted by the confidence mask).
// ---------------------------------------------------------------------------
__global__ __launch_bounds__(TPB) void fine_select(const unsigned* __restrict__ hist2,
                                                   const int* __restrict__ t1,
                                                   const unsigned* __restrict__ kin,
                                                   unsigned* __restrict__ thr,
                                                   unsigned long long* __restrict__ acc) {
  const int bc = blockIdx.x;
  const int tc = t1[bc];
  if (tc < 0) { if (threadIdx.x == 0) thr[bc] = 0xFFFFFFFFu; return; }   // k==0
  const unsigned* h = hist2 + (size_t)bc * NBINS2;
  __shared__ unsigned csum[TPB];
  constexpr int CH = NBINS2 / TPB;   // 16
  const int lo = threadIdx.x * CH;
  unsigned s = 0;
  for (int i = lo; i < lo + CH; ++i) s += h[i];
  csum[threadIdx.x] = s;
  __syncthreads();
  if (threadIdx.x == 0) {
    unsigned k = kin[bc];
    unsigned cum = 0; int ch = TPB - 1;
    while (cum + csum[ch] < k) { cum += csum[ch]; --ch; }
    int bin = ch * CH + CH - 1;
    while (cum + h[bin] < k) { cum += h[bin]; --bin; }
    const unsigned rEq = k - cum;                         // ties to take
    const unsigned T = PBASE + ((unsigned)tc << 12) + (unsigned)bin;
    thr[bc] = T;
    const float pt = __uint_as_float(T);
    if (!(pt > 0.9f)) {
      double contrib = (double)rEq * (double)(-__logf(pt));
      atomicAdd(acc, (unsigned long long)(contrib * FIXSCALE + 0.5));
    }
  }
}

// ---------------------------------------------------------------------------
// Kernel E: masked nll sum. include = (p > 0.9) || (bits > T). Equal-to-T
// pixels were handled analytically in kernel D. Deterministic integer
// fixed-point atomic per block.
// ---------------------------------------------------------------------------
__global__ __launch_bounds__(TPB) void loss_kernel(const float* __restrict__ maxp,
                                                   const float* __restrict__ nllp,
                                                   const unsigned* __restrict__ pseudo4,
                                                   const unsigned* __restrict__ thr,
                                                   unsigned long long* __restrict__ acc) {
  const int gp = (blockIdx.x * TPB + threadIdx.x) * 4;
  const int b  = gp >> 19;
  float4 p = *(const float4*)&maxp[gp];
  float4 l = *(const float4*)&nllp[gp];
  const unsigned ps = pseudo4[gp >> 2];
  float pv[4] = {p.x, p.y, p.z, p.w};
  float lv[4] = {l.x, l.y, l.z, l.w};
  float local = 0.0f;
#pragma unroll
  for (int j = 0; j < 4; ++j) {
    unsigned u = __float_as_uint(pv[j]);
    unsigned T = thr[b * C_ + ((ps >> (8 * j)) & 0xFF)];
    if (pv[j] > 0.9f || u > T) local += lv[j];
  }
  __shared__ float red[TPB];
  red[threadIdx.x] = local;
  __syncthreads();
  for (int s2 = TPB / 2; s2 > 0; s2 >>= 1) {
    if (threadIdx.x < s2) red[threadIdx.x] += red[threadIdx.x + s2];
    __syncthreads();
  }
  if (threadIdx.x == 0)
    atomicAdd(acc, (unsigned long long)((double)red[0] * FIXSCALE + 0.5));
}

__global__ void finalize_kernel(const unsigned long long* __restrict__ acc,
                                float* __restrict__ out) {
  out[0] = (float)((double)(*acc) * (1.0 / FIXSCALE) / (double)TOTAL_);
}

// ---------------------------------------------------------------------------
extern "C" void kernel_launch(void* const* d_in, const int* in_sizes, int n_in,
                              void* d_out, int out_size, void* d_ws, size_t ws_size,
                              hipStream_t stream) {
  (void)in_sizes; (void)n_in; (void)out_size; (void)ws_size;
  const float* pred = (const float*)d_in[0];
  char* ws = (char*)d_ws;
  auto acc   = (unsigned long long*)(ws + OFF_ACC);
  auto hist1 = (unsigned*)(ws + OFF_HIST1);
  auto hist2 = (unsigned*)(ws + OFF_HIST2);
  auto t1    = (int*)     (ws + OFF_T1);
  auto kin   = (unsigned*)(ws + OFF_KIN);
  auto thr   = (unsigned*)(ws + OFF_THR);
  auto maxp  = (float*)   (ws + OFF_MAXP);
  auto nllp  = (float*)   (ws + OFF_NLL);
  auto ps4   = (unsigned*)(ws + OFF_PS);

  hipMemsetAsync(ws, 0, ZERO_BYTES, stream);   // acc + both histograms
  stats_kernel   <<<NBLK_PIX, TPB, 0, stream>>>(pred, maxp, nllp, ps4, hist1);
  coarse_select  <<<BC_,      TPB, 0, stream>>>(hist1, t1, kin);
  fine_hist      <<<NBLK_PIX, TPB, 0, stream>>>(maxp, ps4, t1, hist2);
  fine_select    <<<BC_,      TPB, 0, stream>>>(hist2, t1, kin, thr, acc);
  loss_kernel    <<<NBLK_PIX, TPB, 0, stream>>>(maxp, nllp, ps4, thr, acc);
  finalize_kernel<<<1, 1, 0, stream>>>(acc, (float*)d_out);
}